// AllPassMORRCirculantConv2d_24412594111223
// MI455X (gfx1250) — compile-verified
//
#include <hip/hip_runtime.h>

typedef __attribute__((ext_vector_type(2))) float v2f;
typedef __attribute__((ext_vector_type(4))) float v4f;
typedef __attribute__((ext_vector_type(8))) float v8f;

#define A_MRR 0.987f
#define R_MRR 0.99f

// Block = 256 threads (8 wave32), owns 32 consecutive ow pixels of one (b, oh) row.
// Wave w: ptile = w>>2 (which 16-pixel tile), ctile = w&3 (which 16 of 64 output cols).
// Per q (36 iters): phase[16x16] = A[16x8] @ Bq[8x16] via two chained
// v_wmma_f32_16x16x4_f32, then elementwise MRR transfer + scaled accumulate.
__launch_bounds__(256)
__global__ void morr_circconv_kernel(const float* __restrict__ x,
                                     const float* __restrict__ w,
                                     const float* __restrict__ sc_in,
                                     float* __restrict__ out)
{
    __shared__ float patch[32 * 3 * 36];   // squared input patch [c][kh][36], wl=0..33
    __shared__ float wabs[8 * 36 * 8];     // |w| [p][q][k]
    __shared__ float scale[36];            // +s[0:18], -s[0:18]
    __shared__ int   offTab[288];          // channel -> patch base offset (c*108+kh*36+kw)
    __shared__ float outBuf[64 * 36];      // [col][pixel] staging (stride 36 -> 144B, 16B aligned)

    const int tid = threadIdx.x;
    const int wg  = blockIdx.x;            // 1024 = 8 b * 64 oh * 2 halves
    const int b   = wg >> 7;
    const int rem = wg & 127;
    const int oh  = rem >> 1;
    const int ow0 = (rem & 1) << 5;

    // differential-rail scale (q=36 even branch): concat(s[:18], -s[:18])
    if (tid < 18) {
        float s = sc_in[tid];
        scale[tid]      = s;
        scale[tid + 18] = -s;
    }
    // channel -> (c, kh, kw) patch offset
    for (int ch = tid; ch < 288; ch += 256) {
        unsigned c  = (unsigned)ch / 9u;
        unsigned r  = (unsigned)ch - 9u * c;
        unsigned kh = r / 3u;
        unsigned kw = r - 3u * kh;
        offTab[ch] = (int)(c * 108u + kh * 36u + kw);
    }
    // build_weight() = |W|
    for (int i = tid; i < 2304; i += 256) wabs[i] = fabsf(w[i]);
    // squared, zero-padded input patch: rows oh-1..oh+1, cols ow0-1..ow0+32
    for (int e = tid; e < 32 * 3 * 34; e += 256) {
        int c   = e / 102;
        int r2  = e - c * 102;
        int kh  = r2 / 34;
        int wl  = r2 - kh * 34;
        int row = oh + kh - 1;
        int col = ow0 + wl - 1;
        float v = 0.0f;
        if ((unsigned)row < 64u && (unsigned)col < 64u)
            v = x[(((b * 32 + c) * 64) + row) * 64 + col];
        patch[c * 108 + kh * 36 + wl] = v * v;   // intensity modulation
    }
    __syncthreads();

    const int lane  = tid & 31;
    const int wid   = tid >> 5;
    const int ptile = wid >> 2;    // 0..1
    const int ctile = wid & 3;     // 0..3
    const int ln    = lane & 15;
    const int hi    = lane >> 4;   // K-half select per ISA A/B fragment layout

    const int pixA = ptile * 16 + ln;         // A row (pixel within workgroup)
    const int colB = ctile * 16 + ln;         // B col (global output column p*8+t)
    const int pB   = colB >> 3;
    const int tB   = colB & 7;

    // K slots held by this lane: s = {2hi, 2hi+1} (frag0 vgpr0/1), {2hi+4, 2hi+5} (frag1)
    const int s0 = 2 * hi;
    // q-invariant circular-conv rotations for B: B[s, col] = |w[p, q, (t-s)&7]|
    const int rot0 = (tB - (s0    )) & 7;
    const int rot1 = (tB - (s0 + 1)) & 7;
    const int rot2 = (tB - (s0 + 4)) & 7;
    const int rot3 = (tB - (s0 + 5)) & 7;
    const float* wb = &wabs[pB * 288];        // p * 36 * 8

    const float twoAR  = 2.0f * A_MRR * R_MRR;
    const float numc   = A_MRR * A_MRR + R_MRR * R_MRR;
    const float denc   = 1.0f + (A_MRR * R_MRR) * (A_MRR * R_MRR);
    const float inv2pi = 0.15915494309189535f;   // v_cos_f32 takes revolutions

    v8f acc = {};

    for (int q = 0; q < 36; ++q) {
        const int chb = q * 8 + s0;
        v2f a0, a1, b0, b1;
        a0.x = patch[offTab[chb    ] + pixA];
        a0.y = patch[offTab[chb + 1] + pixA];
        a1.x = patch[offTab[chb + 4] + pixA];
        a1.y = patch[offTab[chb + 5] + pixA];
        const float* wq = wb + q * 8;
        b0.x = wq[rot0];
        b0.y = wq[rot1];
        b1.x = wq[rot2];
        b1.y = wq[rot3];

        v8f ph = {};
        ph = __builtin_amdgcn_wmma_f32_16x16x4_f32(false, a0, false, b0,
                                                   (short)0, ph, false, false);
        ph = __builtin_amdgcn_wmma_f32_16x16x4_f32(false, a1, false, b1,
                                                   (short)0, ph, false, false);

        const float scv = scale[q];
        #pragma unroll
        for (int v = 0; v < 8; ++v) {
            float c  = __builtin_amdgcn_cosf(ph[v] * inv2pi);
            float u  = twoAR * c;
            float tr = (numc - u) * __builtin_amdgcn_rcpf(denc - u);
            acc[v] = fmaf(scv, tr, acc[v]);
        }
    }

    // C layout: acc[v] on lane -> row M = v + 8*hi, col N = ln (within tile)
    #pragma unroll
    for (int v = 0; v < 8; ++v) {
        int pix = ptile * 16 + v + 8 * hi;
        outBuf[colB * 36 + pix] = acc[v];
    }
    __syncthreads();

    // coalesced vector stores: out[b, col, oh, ow0 + po .. +3]
    float* obase = out + ((b * 64) * 64 + oh) * 64 + ow0;
    #pragma unroll
    for (int i = 0; i < 2; ++i) {
        int e   = tid + i * 256;     // 512 float4s total
        int col = e >> 3;
        int po  = (e & 7) << 2;
        v4f val = *(const v4f*)&outBuf[col * 36 + po];
        *(v4f*)(obase + col * 4096 + po) = val;
    }
}

extern "C" void kernel_launch(void* const* d_in, const int* in_sizes, int n_in,
                              void* d_out, int out_size, void* d_ws, size_t ws_size,
                              hipStream_t stream) {
    const float* x  = (const float*)d_in[0];   // [8, 32, 64, 64] f32
    const float* w  = (const float*)d_in[1];   // [8, 36, 8] f32
    const float* sc = (const float*)d_in[2];   // [19] f32
    float* out = (float*)d_out;                // [8, 64, 64, 64] f32
    (void)in_sizes; (void)n_in; (void)out_size; (void)d_ws; (void)ws_size;
    morr_circconv_kernel<<<dim3(1024), dim3(256), 0, stream>>>(x, w, sc, out);
}